// GCN_P_1623497638173
// MI455X (gfx1250) — compile-verified
//
#include <hip/hip_runtime.h>

typedef __attribute__((ext_vector_type(2))) float v2f;
typedef __attribute__((ext_vector_type(4))) float v4f;
typedef __attribute__((ext_vector_type(8))) float v8f;
typedef __attribute__((ext_vector_type(4))) unsigned int u32x4;
typedef __attribute__((ext_vector_type(8))) unsigned int u32x8;

#define NN 40000
#define NE 640000
#define CH 128
#define TSTRIDE 132   // padded stride for transposed W: bank = (4n + k) % 64 -> conflict-free
#define BN_EPS 1e-5f

// ---------------- init: zero degree counters + BN accumulators ----------------
__global__ __launch_bounds__(256) void k_init(float* __restrict__ deg,
                                              float* __restrict__ sums,
                                              float* __restrict__ sumsq) {
  int i = blockIdx.x * blockDim.x + threadIdx.x;
  if (i < NN) deg[i] = 0.0f;
  if (i < CH) { sums[i] = 0.0f; sumsq[i] = 0.0f; }
}

// ---------------- degree count over dst ----------------
__global__ __launch_bounds__(256) void k_deg(const int* __restrict__ dst,
                                             float* __restrict__ deg) {
  int e = blockIdx.x * blockDim.x + threadIdx.x;
  if (e < NE) atomicAdd(&deg[dst[e]], 1.0f);
}

// ---------------- dinv = rsqrt(deg + 1) ----------------
__global__ __launch_bounds__(256) void k_dinv(const float* __restrict__ deg,
                                              float* __restrict__ dinv) {
  int i = blockIdx.x * blockDim.x + threadIdx.x;
  if (i < NN) dinv[i] = rsqrtf(deg[i] + 1.0f);
}

// ---------------- h = x @ W via V_WMMA_F32_16X16X4_F32 ----------------
// W DMA'd row-major to LDS by the Tensor Data Mover, then repacked once into a
// transposed padded image sWt[n*132+k] so each B fragment (W[k'][n], W[k'+1][n])
// is one conflict-free ds_load_b64 straight into an even VGPR pair.
__global__ __launch_bounds__(128) void k_gemm(const float* __restrict__ x,
                                              const float* __restrict__ W,
                                              float* __restrict__ h) {
  __shared__ float sW[CH * CH];        // 64 KB row-major TDM landing buffer
  __shared__ float sWt[CH * TSTRIDE];  // 66 KB transposed+padded (total 130 KB < 320 KB)
  const int tid = threadIdx.x;

  // --- TDM: DMA the full 128x128 f32 W tile from global into LDS ---
  if ((tid >> 5) == 0) {  // wave 0 issues (TDM ignores EXEC; duplicate issue is benign)
    unsigned long long wa = (unsigned long long)(uintptr_t)W;
    unsigned int ldsOff = (unsigned int)(uintptr_t)(&sW[0]);  // flat LDS addr[31:0] = LDS offset

    u32x4 g0;
    g0[0] = 1u;                                  // count=1, is_restore=0, gather_mode=0
    g0[1] = ldsOff;                              // lds_addr (bytes)
    g0[2] = (unsigned int)(wa & 0xFFFFFFFFu);    // global_addr[31:0]
    g0[3] = ((unsigned int)(wa >> 32) & 0x01FFFFFFu) | 0x80000000u;  // addr[56:32] | type=2

    u32x8 g1;
    g1[0] = 0x00020000u;   // workgroup_mask=0 (not in cluster), data_size=2 (4 bytes)
    g1[1] = 128u << 16;    // tensor_dim0[15:0] = 128   (bits 63:48)
    g1[2] = 128u << 16;    // tensor_dim1[15:0] = 128   (bits 95:80); dim0[31:16]=0
    g1[3] = 128u << 16;    // tile_dim0 = 128           (bits 127:112); dim1[31:16]=0
    g1[4] = 128u;          // tile_dim1 = 128           (bits 143:128); tile_dim2=0 -> 2D
    g1[5] = 128u;          // tensor_dim0_stride[31:0] = 128 elements
    g1[6] = 0u;            // stride[47:32]=0; tensor_dim1_stride[15:0]=0 (unused, 2D)
    g1[7] = 0u;            // tensor_dim1_stride[47:16]=0

    asm volatile("tensor_load_to_lds %0, %1" :: "s"(g0), "s"(g1) : "memory");
    __builtin_amdgcn_s_wait_tensorcnt(0);
  }
  __syncthreads();

  // --- one-time repack: thread t owns column n=t; sWt[n][k] = W[k][n] ---
  {
    const int n = tid;  // 0..127
#pragma unroll 4
    for (int k = 0; k < CH; k += 4) {
      v4f w;
      w.x = sW[(k + 0) * CH + n];
      w.y = sW[(k + 1) * CH + n];
      w.z = sW[(k + 2) * CH + n];
      w.w = sW[(k + 3) * CH + n];
      *(v4f*)(&sWt[n * TSTRIDE + k]) = w;  // 16B aligned
    }
  }
  __syncthreads();

  const int wave  = tid >> 5;
  const int lane  = tid & 31;
  const int mBase = blockIdx.x * 64 + wave * 16;
  const int mRow  = mBase + (lane & 15);
  const int kSel  = (lane >> 4) << 1;  // 0 (lanes 0-15) or 2 (lanes 16-31)
  const int nLane = lane & 15;

  v8f acc[8];
#pragma unroll
  for (int nt = 0; nt < 8; ++nt) acc[nt] = (v8f){0.f,0.f,0.f,0.f,0.f,0.f,0.f,0.f};

  const float* xrow = x + (size_t)mRow * CH;
  for (int k = 0; k < CH; k += 4) {
    v2f a = *(const v2f*)(xrow + k + kSel);  // one global_load_b64 (8B aligned)
#pragma unroll
    for (int nt = 0; nt < 8; ++nt) {
      const int n = nt * 16 + nLane;
      v2f b = *(const v2f*)(sWt + n * TSTRIDE + k + kSel);  // one ds_load_b64, no conflicts
      // (neg_a, A, neg_b, B, c_mod, C, reuse_a, reuse_b)
      acc[nt] = __builtin_amdgcn_wmma_f32_16x16x4_f32(
          false, a, false, b, (short)0, acc[nt], false, false);
    }
  }

  const int mOff = (lane >> 4) << 3;  // 0 or 8
#pragma unroll
  for (int nt = 0; nt < 8; ++nt) {
#pragma unroll
    for (int i = 0; i < 8; ++i) {
      h[(size_t)(mBase + mOff + i) * CH + nt * 16 + nLane] = acc[nt][i];
    }
  }
}

// ---------------- agg init = self-loop contribution h * dinv^2 = h / (deg + 1) ----------------
__global__ __launch_bounds__(256) void k_selfloop(const float* __restrict__ h,
                                                  const float* __restrict__ dinv,
                                                  float* __restrict__ agg) {
  int idx = blockIdx.x * blockDim.x + threadIdx.x;  // exact cover of NN*CH
  int node = idx >> 7;
  float di = dinv[node];
  agg[idx] = h[idx] * di * di;
}

// ---------------- edge scatter: agg[dst] += h[src] * dinv[src]*dinv[dst] ----------------
// 32 lanes per edge, 4 channels per lane (float4 gather, 4 scalar f32 atomics).
__global__ __launch_bounds__(256) void k_edges(const int* __restrict__ src,
                                               const int* __restrict__ dst,
                                               const float* __restrict__ dinv,
                                               const float* __restrict__ h,
                                               float* __restrict__ agg) {
  int gid = blockIdx.x * blockDim.x + threadIdx.x;  // exact cover of NE*32
  int e = gid >> 5;
  int l = gid & 31;
  int s = src[e], d = dst[e];
  float norm = dinv[s] * dinv[d];
  v4f hv = *(const v4f*)(h + (size_t)s * CH + l * 4);
  float* ap = agg + (size_t)d * CH + l * 4;
  atomicAdd(ap + 0, hv.x * norm);
  atomicAdd(ap + 1, hv.y * norm);
  atomicAdd(ap + 2, hv.z * norm);
  atomicAdd(ap + 3, hv.w * norm);
}

// ---------------- per-channel sum / sumsq (bias b cancels in BN, omitted) ----------------
__global__ __launch_bounds__(256) void k_colstats(const float* __restrict__ agg,
                                                  float* __restrict__ sums,
                                                  float* __restrict__ sumsq) {
  __shared__ float rs[256], rq[256];
  const int tid  = threadIdx.x;
  const int c    = tid & 127;
  const int half = tid >> 7;
  const int r0   = blockIdx.x * 160;  // 250 blocks * 160 rows == 40000
  float s = 0.0f, q = 0.0f;
  for (int r = r0 + half; r < r0 + 160; r += 2) {
    float v = agg[(size_t)r * CH + c];
    s += v;
    q += v * v;
  }
  rs[tid] = s; rq[tid] = q;
  __syncthreads();
  if (half == 0) {
    s += rs[tid + 128];
    q += rq[tid + 128];
    atomicAdd(&sums[c], s);
    atomicAdd(&sumsq[c], q);
  }
}

// ---------------- fused BatchNorm (batch stats) + ReLU ----------------
__global__ __launch_bounds__(256) void k_bnrelu(const float* __restrict__ agg,
                                                const float* __restrict__ sums,
                                                const float* __restrict__ sumsq,
                                                const float* __restrict__ gamma,
                                                const float* __restrict__ beta,
                                                float* __restrict__ out) {
  int idx = blockIdx.x * blockDim.x + threadIdx.x;  // exact cover of NN*CH
  int c = idx & 127;
  const float invN = 1.0f / (float)NN;
  float mean = sums[c] * invN;
  float var  = sumsq[c] * invN - mean * mean;  // biased variance == E[x^2]-E[x]^2
  float v = gamma[c] * (agg[idx] - mean) * rsqrtf(var + BN_EPS) + beta[c];
  out[idx] = v > 0.0f ? v : 0.0f;
}

extern "C" void kernel_launch(void* const* d_in, const int* in_sizes, int n_in,
                              void* d_out, int out_size, void* d_ws, size_t ws_size,
                              hipStream_t stream) {
  (void)in_sizes; (void)n_in; (void)out_size; (void)ws_size;

  const float* x     = (const float*)d_in[0];
  const int*   ei    = (const int*)d_in[1];
  const float* W     = (const float*)d_in[2];
  /* d_in[3] = b : provably a no-op through BatchNorm (per-channel constant) */
  const float* gamma = (const float*)d_in[4];
  const float* beta  = (const float*)d_in[5];
  float* out = (float*)d_out;

  const int* src = ei;
  const int* dst = ei + NE;

  char* ws = (char*)d_ws;
  float* h     = (float*)(ws);                                    // NN*CH f32
  float* agg   = (float*)(ws + (size_t)NN * CH * 4);              // NN*CH f32
  float* deg   = (float*)(ws + (size_t)NN * CH * 8);              // NN f32
  float* dinv  = (float*)(ws + (size_t)NN * CH * 8 + (size_t)NN * 4);
  float* sums  = (float*)(ws + (size_t)NN * CH * 8 + (size_t)NN * 8);
  float* sumsq = sums + CH;

  k_init    <<<(NN + 255) / 256, 256, 0, stream>>>(deg, sums, sumsq);
  k_deg     <<<NE / 256,         256, 0, stream>>>(dst, deg);
  k_dinv    <<<(NN + 255) / 256, 256, 0, stream>>>(deg, dinv);
  k_gemm    <<<NN / 64,          128, 0, stream>>>(x, W, h);
  k_selfloop<<<NN * CH / 256,    256, 0, stream>>>(h, dinv, agg);
  k_edges   <<<NE / 8,           256, 0, stream>>>(src, dst, dinv, h, agg);
  k_colstats<<<NN / 160,         256, 0, stream>>>(agg, sums, sumsq);
  k_bnrelu  <<<NN * CH / 256,    256, 0, stream>>>(agg, sums, sumsq, gamma, beta, out);
}